// QuantizedEmbedding_15118284882612
// MI455X (gfx1250) — compile-verified
//
#include <hip/hip_runtime.h>

typedef __attribute__((ext_vector_type(2))) float v2f;
typedef __attribute__((ext_vector_type(8))) float v8f;
typedef __attribute__((ext_vector_type(4))) int   v4i;

#define N_PTS   65536
#define K_CODES 1024
#define D_DIM   256
#define GAMMA   0.99f

// ---- workspace layout (float units) ----
#define WS_WSQ  0                               // [K]   ||w||^2
#define WS_NT   (K_CODES)                       // [K]   one-hot column sums
#define WS_ST   (2 * K_CODES)                   // [K*D] segment sums of ze
#define WS_DIST (2 * K_CODES + K_CODES * D_DIM) // [1]   sum of squared diffs
#define WS_IDX  (WS_DIST + 1)                   // [N]   int32 argmin indices

// ---- d_out layout (flat, reference return order) ----
#define OUT_ZQ   0
#define OUT_DIST (N_PTS * D_DIM)
#define OUT_IDX  (OUT_DIST + 1)
#define OUT_EMB  (OUT_IDX + N_PTS)
#define OUT_MT   (OUT_EMB + K_CODES * D_DIM)
#define OUT_NT   (OUT_MT + K_CODES * D_DIM)

#define W_PITCH 260   // LDS pitch: 260 % 64 banks = 4 -> conflict-free column reads,
                      // 260*4 bytes is 16B-aligned so float4/b128 staging stays aligned

#if defined(__has_builtin)
#if __has_builtin(__builtin_amdgcn_global_load_async_to_lds_b128)
#define HAVE_ASYNC_LDS 1
#endif
#endif
#ifndef HAVE_ASYNC_LDS
#define HAVE_ASYNC_LDS 0
#endif

static __device__ __forceinline__ void wait_asynccnt0() {
#if defined(__has_builtin) && __has_builtin(__builtin_amdgcn_s_wait_asynccnt)
    __builtin_amdgcn_s_wait_asynccnt(0);
#else
    asm volatile("s_wait_asynccnt 0" ::: "memory");
#endif
}

// Zero the EMA accumulators + dist each call (harness only poisons once).
__global__ void init_ws(float* __restrict__ ws) {
    int i = blockIdx.x * blockDim.x + threadIdx.x;
    const int total = K_CODES + K_CODES * D_DIM + 1; // nt + st + dist
    if (i < total) ws[WS_NT + i] = 0.0f;
}

// ||w_k||^2 for each code row.
__global__ void wsq_kernel(const float* __restrict__ W, float* __restrict__ ws) {
    int k = blockIdx.x * blockDim.x + threadIdx.x;
    if (k >= K_CODES) return;
    const float4* row = (const float4*)(W + (size_t)k * D_DIM);
    float s = 0.0f;
#pragma unroll 8
    for (int i = 0; i < D_DIM / 4; ++i) {
        float4 v = row[i];
        s += v.x * v.x + v.y * v.y + v.z * v.z + v.w * v.w;
    }
    ws[WS_WSQ + k] = s;
}

// Fused distance GEMM (f32 WMMA 16x16x4) + per-row argmin over K.
// Block = 256 threads = 8 waves; each wave owns a 16-row tile of ze (128 rows/block).
// W tiles are double-buffered in LDS via GLOBAL_LOAD_ASYNC_TO_LDS_B128 (ASYNCcnt),
// so the prefetch of tile ct+1 overlaps the 64 WMMAs of tile ct.
__launch_bounds__(256)
__global__ void argmin_wmma(const float* __restrict__ ze,
                            const float* __restrict__ W,
                            const float* __restrict__ wsq,
                            float* __restrict__ out_idx_f,
                            int* __restrict__ out_idx_i) {
#if HAVE_ASYNC_LDS
    __shared__ float Wl[2][16 * W_PITCH];
#else
    __shared__ float Wl[1][16 * W_PITCH];
#endif

    const int lane    = threadIdx.x & 31;
    const int wave    = threadIdx.x >> 5;
    const int rowBase = blockIdx.x * 128 + wave * 16;
    const int col     = lane & 15;            // code column within tile / A-row index
    const int hiOff   = (lane >= 16) ? 2 : 0; // K-offset for A/B fragments (ISA layout)

    float bestVal[8];
    int   bestIdx[8];
#pragma unroll
    for (int v = 0; v < 8; ++v) { bestVal[v] = 3.4e38f; bestIdx[v] = 0; }

    // A fragment source: row M = rowBase+col, K(d) offsets {hiOff, hiOff+1}
    const float* zrow = ze + (size_t)(rowBase + col) * D_DIM + hiOff;

#if HAVE_ASYNC_LDS
    // async DMA of one 16x256 W tile (1024 float4s, 4 b128 ops per thread)
    auto stage_async = [&](int ct, int buf) {
        const float* src = W + (size_t)ct * 16 * D_DIM;
#pragma unroll
        for (int fi = threadIdx.x; fi < 16 * 64; fi += 256) {
            int r = fi >> 6, d4 = fi & 63;
            __builtin_amdgcn_global_load_async_to_lds_b128(
                (__attribute__((address_space(1))) v4i*)(src + fi * 4),
                (__attribute__((address_space(3))) v4i*)&Wl[buf][r * W_PITCH + d4 * 4],
                0, 0);
        }
    };
    stage_async(0, 0);
#endif

    for (int ct = 0; ct < K_CODES / 16; ++ct) {
        const int codeBase = ct * 16;
#if HAVE_ASYNC_LDS
        const int buf = ct & 1;
        wait_asynccnt0();  // my share of tile ct is in LDS
        __syncthreads();   // whole tile in LDS; prior reads of other buffer done
        if (ct + 1 < K_CODES / 16) stage_async(ct + 1, buf ^ 1);
#else
        const int buf = 0;
        __syncthreads(); // previous tile's LDS reads complete
        {
            const float4* src = (const float4*)(W + (size_t)codeBase * D_DIM);
#pragma unroll
            for (int fi = threadIdx.x; fi < 16 * 64; fi += 256) {
                int r = fi >> 6, d4 = fi & 63;
                *(float4*)&Wl[0][r * W_PITCH + d4 * 4] = src[fi];
            }
        }
        __syncthreads();
#endif

        v8f acc = {};
#pragma unroll 4
        for (int d0 = 0; d0 < D_DIM; d0 += 4) {
            v2f a = *(const v2f*)(zrow + d0);
            v2f b = *(const v2f*)&Wl[buf][col * W_PITCH + d0 + hiOff];
            acc = __builtin_amdgcn_wmma_f32_16x16x4_f32(
                false, a, false, b, (short)0, acc, false, false);
        }

        const float wsq_c = wsq[codeBase + col];
        const int   myIdx = codeBase + col;
        // Branchless per-lane running min. Strict '<' keeps the first (lowest
        // index) minimum within this lane's increasing column sequence.
#pragma unroll
        for (int v = 0; v < 8; ++v) {
            float val = fmaf(-2.0f, acc[v], wsq_c);
            bool  t   = val < bestVal[v];
            bestVal[v] = t ? val : bestVal[v];
            bestIdx[v] = t ? myIdx : bestIdx[v];
        }
    }

    // One final branchless butterfly argmin across each 16-lane group.
#pragma unroll
    for (int v = 0; v < 8; ++v) {
        float val = bestVal[v];
        int   idx = bestIdx[v];
#pragma unroll
        for (int off = 8; off >= 1; off >>= 1) {
            float ov = __shfl_xor(val, off, 32);
            int   oi = __shfl_xor(idx, off, 32);
            bool  t  = (ov < val) | ((ov == val) & (oi < idx));
            val = t ? ov : val;
            idx = t ? oi : idx;
        }
        bestVal[v] = val;
        bestIdx[v] = idx;
    }

    // C/D layout: accumulator v holds row M=v (lanes 0-15) or M=v+8 (lanes 16-31)
    if (lane == 0 || lane == 16) {
        int rbase = rowBase + ((lane >= 16) ? 8 : 0);
#pragma unroll
        for (int v = 0; v < 8; ++v) {
            out_idx_i[rbase + v] = bestIdx[v];
            out_idx_f[rbase + v] = (float)bestIdx[v];
        }
    }
}

// Gather zq with OLD codebook, straight-through output, commitment loss,
// and EMA segment sums (nt, st) via float atomics.
__launch_bounds__(256)
__global__ void gather_stats(const float* __restrict__ ze,
                             const float* __restrict__ W,
                             const int* __restrict__ idx,
                             float* __restrict__ out_zq,
                             float* __restrict__ st,
                             float* __restrict__ nt,
                             float* __restrict__ distAcc) {
    __shared__ float red[256];
    const int n = blockIdx.x;
    const int d = threadIdx.x;
    const int k = idx[n];

    float z = ze[(size_t)n * D_DIM + d];
    float q = W[(size_t)k * D_DIM + d];
    float diff = q - z;
    out_zq[(size_t)n * D_DIM + d] = z + diff; // ze + (zq - ze), matches JAX rounding

    atomicAdd(&st[(size_t)k * D_DIM + d], z);
    if (d == 0) atomicAdd(&nt[k], 1.0f);

    red[d] = diff * diff;
    __syncthreads();
#pragma unroll
    for (int s = 128; s > 0; s >>= 1) {
        if (d < s) red[d] += red[d + s];
        __syncthreads();
    }
    if (d == 0) atomicAdd(distAcc, red[0]);
}

// EMA update, codebook division, loss mean.
__global__ void finalize_kernel(const float* __restrict__ mt,
                                const float* __restrict__ Nt,
                                const float* __restrict__ ws,
                                float* __restrict__ out) {
    int id = blockIdx.x * blockDim.x + threadIdx.x;
    if (id == 0)
        out[OUT_DIST] = ws[WS_DIST] / (float)(N_PTS * D_DIM);
    if (id >= K_CODES * D_DIM) return;
    int k = id / D_DIM, d = id % D_DIM;
    float NtN = GAMMA * Nt[k] + (1.0f - GAMMA) * ws[WS_NT + k];
    float mtN = GAMMA * mt[id] + (1.0f - GAMMA) * ws[WS_ST + id];
    out[OUT_EMB + id] = mtN / NtN;
    out[OUT_MT + id]  = mtN;
    if (d == 0) out[OUT_NT + k] = NtN;
}

extern "C" void kernel_launch(void* const* d_in, const int* in_sizes, int n_in,
                              void* d_out, int out_size, void* d_ws, size_t ws_size,
                              hipStream_t stream) {
    const float* ze     = (const float*)d_in[0];
    const float* embedW = (const float*)d_in[1];
    const float* mt     = (const float*)d_in[2];
    const float* Nt     = (const float*)d_in[3];
    float* out = (float*)d_out;
    float* ws  = (float*)d_ws;
    int*   wsIdx = (int*)(ws + WS_IDX);

    // 1. zero EMA accumulators + dist
    {
        int total = K_CODES + K_CODES * D_DIM + 1;
        init_ws<<<(total + 255) / 256, 256, 0, stream>>>(ws);
    }
    // 2. ||w||^2
    wsq_kernel<<<(K_CODES + 255) / 256, 256, 0, stream>>>(embedW, ws + WS_WSQ);
    // 3. fused distance GEMM + argmin (f32 WMMA, async double-buffered W tiles)
    argmin_wmma<<<N_PTS / 128, 256, 0, stream>>>(ze, embedW, ws + WS_WSQ,
                                                 out + OUT_IDX, wsIdx);
    // 4. gather + straight-through + loss + segment sums
    gather_stats<<<N_PTS, 256, 0, stream>>>(ze, embedW, wsIdx, out + OUT_ZQ,
                                            ws + WS_ST, ws + WS_NT, ws + WS_DIST);
    // 5. EMA finalize
    finalize_kernel<<<(K_CODES * D_DIM + 255) / 256, 256, 0, stream>>>(mt, Nt, ws, out);
}